// EmNet_16389595202342
// MI455X (gfx1250) — compile-verified
//
#include <hip/hip_runtime.h>

#define VOCABN 50257
#define EMB    100
#define STACKS 3
#define TSTEPS 15
#define BATCH  2048
#define ROWS   (BATCH * TSTEPS)   // 30720
#define GDIM   (4 * EMB)          // 400

// LDS layout constants for k_recurrent (dynamic shared memory, 320KB/WGP on gfx1250)
#define WSTR   112                // weight row stride: +2 rows = +224 dw = +32 banks -> conflict-free
#define WGATE  (EMB * WSTR)       // 11200 floats per gate
#define GSTR   113                // gate/h row stride (odd -> conflict-free)
#define GGATE  (16 * GSTR)        // 1808 floats per gate tile
#define SMEM_FLOATS (4 * WGATE + 4 * GGATE + 16 * GSTR)   // 53840 floats = 215360 B

#define NOLIMIT (1 << 30)         // fold-away bound for interior tiles

typedef float v2f __attribute__((ext_vector_type(2)));
typedef float v8f __attribute__((ext_vector_type(8)));

static __device__ __forceinline__ v8f wmma_f32(v2f a, v2f b, v8f c) {
    // V_WMMA_F32_16X16X4_F32 : D = A(16x4) * B(4x16) + C(16x16)
    return __builtin_amdgcn_wmma_f32_16x16x4_f32(
        /*neg_a=*/false, a, /*neg_b=*/false, b,
        /*c_mod=*/(short)0, c, /*reuse_a=*/false, /*reuse_b=*/false);
}

static __device__ __forceinline__ float sigmoidf_(float x) {
    return 1.0f / (1.0f + __expf(-x));
}

// kc-outer GEMM core: per kc, one clause of 2*NT B-loads feeds 2*NT WMMAs.
template <int NT>
static __device__ __forceinline__ void gemm_tiles(
    const v2f a0[25], const v2f a1[25],
    const float* __restrict__ W, size_t ld, int nstart,
    int l16, int half, int nlimit,
    v8f acc0[NT], v8f acc1[NT]) {
#pragma unroll
    for (int kc = 0; kc < 25; ++kc) {
        const int k0 = kc * 4 + 2 * half;
        v2f b[NT];
#pragma unroll
        for (int nt = 0; nt < NT; ++nt) {
            int n = nstart + nt * 16 + l16;
            bool ok = n < nlimit;
            b[nt].x = ok ? W[(size_t)(k0 + 0) * ld + n] : 0.0f;
            b[nt].y = ok ? W[(size_t)(k0 + 1) * ld + n] : 0.0f;
        }
#pragma unroll
        for (int nt = 0; nt < NT; ++nt) {
            acc0[nt] = wmma_f32(a0[kc], b[nt], acc0[nt]);
            acc1[nt] = wmma_f32(a1[kc], b[nt], acc1[nt]);
        }
    }
}

// ---------------------------------------------------------------- embedding
__global__ __launch_bounds__(256) void k_embed(const int* __restrict__ x,
                                               const float* __restrict__ emb,
                                               float* __restrict__ hseq) {
    int i = blockIdx.x * 256 + threadIdx.x;     // over ROWS*EMB
    if (i >= ROWS * EMB) return;
    int row = i / EMB;
    int e   = i - row * EMB;
    int tok = x[row];
    hseq[i] = emb[tok * EMB + e];
}

// ------------------------------------------- xpart = hseq @ W*x + bias (GEMM)
// M = ROWS (30720), K = 100, N = 4 gates x 100.  One wave per gate,
// 32 M-rows per block, kc-outer with 4+3 N-tiles live.
__global__ __launch_bounds__(128) void k_xgemm(const float* __restrict__ hseq,
    const float* __restrict__ Wfx, const float* __restrict__ Wix,
    const float* __restrict__ Wcx, const float* __restrict__ Wox,
    const float* __restrict__ bf,  const float* __restrict__ bi,
    const float* __restrict__ bc,  const float* __restrict__ bo,
    float* __restrict__ xpart, int layer) {
    const int m0   = blockIdx.x * 32;
    const int wave = threadIdx.x >> 5;
    const int lane = threadIdx.x & 31;
    const int half = lane >> 4;
    const int l16  = lane & 15;
    const float* W = (wave == 0 ? Wfx : wave == 1 ? Wix : wave == 2 ? Wcx : Wox)
                     + layer * EMB * EMB;
    const float* bias = (wave == 0 ? bf : wave == 1 ? bi : wave == 2 ? bc : bo)
                     + layer * EMB;

    // Preload A fragments for both 16-row M-tiles (K = 25 chunks of 4).
    v2f a0[25], a1[25];
#pragma unroll
    for (int kc = 0; kc < 25; ++kc) {
        const float* p0 = hseq + (size_t)(m0 + l16) * EMB + kc * 4 + 2 * half;
        const float* p1 = p0 + 16 * EMB;
        a0[kc].x = p0[0]; a0[kc].y = p0[1];
        a1[kc].x = p1[0]; a1[kc].y = p1[1];
    }

    // Pass 1: N-tiles 0..3 (n <= 63, fully interior -> guards fold away)
    {
        v8f acc0[4] = {}, acc1[4] = {};
        gemm_tiles<4>(a0, a1, W, EMB, 0, l16, half, NOLIMIT, acc0, acc1);
#pragma unroll
        for (int nt = 0; nt < 4; ++nt) {
            int n = nt * 16 + l16;
            float bb = bias[n];
#pragma unroll
            for (int r = 0; r < 8; ++r) {
                int row = m0 + r + 8 * half;
                xpart[(size_t)row * GDIM + wave * EMB + n]        = acc0[nt][r] + bb;
                xpart[(size_t)(row + 16) * GDIM + wave * EMB + n] = acc1[nt][r] + bb;
            }
        }
    }
    // Pass 2: N-tiles 4..6 (edge at n = 100)
    {
        v8f acc0[3] = {}, acc1[3] = {};
        gemm_tiles<3>(a0, a1, W, EMB, 64, l16, half, EMB, acc0, acc1);
#pragma unroll
        for (int nt = 0; nt < 3; ++nt) {
            int n = 64 + nt * 16 + l16;
            if (n < EMB) {
                float bb = bias[n];
#pragma unroll
                for (int r = 0; r < 8; ++r) {
                    int row = m0 + r + 8 * half;
                    xpart[(size_t)row * GDIM + wave * EMB + n]        = acc0[nt][r] + bb;
                    xpart[(size_t)(row + 16) * GDIM + wave * EMB + n] = acc1[nt][r] + bb;
                }
            }
        }
    }
}

// ------------------------------------------------ recurrent phase-1 core
// kc-outer: one LDS A-load per kc reused across all of this wave's N-tiles.
template <int NT>
static __device__ __forceinline__ void rec_phase1(
    const float* __restrict__ xpart, const float* __restrict__ Wsg,
    const float* __restrict__ h_s, float* __restrict__ gate_s_g,
    int nbase0, int b0, int t, int gate, int l16, int half) {
    v8f acc[NT];
#pragma unroll
    for (int q = 0; q < NT; ++q) {
        int n = nbase0 + q * 16 + l16;
        bool ok = n < EMB;
#pragma unroll
        for (int r = 0; r < 8; ++r) {
            int row = b0 + r + 8 * half;
            acc[q][r] = ok ? xpart[((size_t)row * TSTEPS + t) * GDIM + gate * EMB + n]
                           : 0.0f;
        }
    }
#pragma unroll
    for (int kc = 0; kc < 25; ++kc) {
        const int k0 = kc * 4 + 2 * half;
        v2f a;
        a.x = h_s[l16 * GSTR + k0];
        a.y = h_s[l16 * GSTR + k0 + 1];
#pragma unroll
        for (int q = 0; q < NT; ++q) {
            int n = nbase0 + q * 16 + l16;   // weights zero-padded: no guard
            v2f b;
            b.x = Wsg[(k0 + 0) * WSTR + n];
            b.y = Wsg[(k0 + 1) * WSTR + n];
            acc[q] = wmma_f32(a, b, acc[q]);
        }
    }
#pragma unroll
    for (int q = 0; q < NT; ++q) {
        int n0 = nbase0 + q * 16;
#pragma unroll
        for (int r = 0; r < 8; ++r)
            gate_s_g[(r + 8 * half) * GSTR + n0 + l16] = acc[q][r];
    }
}

// One block = 16 batch rows, all 15 timesteps.  8 waves: wave w handles
// gate w>>1, N-tiles 0..3 (even wave) or 4..6 (odd wave).  All 4 gate weight
// matrices staged once into LDS (zero-padded, stride 112 -> conflict-free).
__global__ __launch_bounds__(256) void k_recurrent(const float* __restrict__ xpart,
    const float* __restrict__ Wfh, const float* __restrict__ Wih,
    const float* __restrict__ Wch, const float* __restrict__ Woh,
    float* __restrict__ hseq, int layer) {
    extern __shared__ float smem[];
    float* Ws     = smem;                         // 4 * EMB * WSTR
    float* gate_s = smem + 4 * WGATE;             // 4 * 16 * GSTR
    float* h_s    = gate_s + 4 * GGATE;           // 16 * GSTR

    const int b0   = blockIdx.x * 16;
    const int tid  = threadIdx.x;
    const int wave = tid >> 5;
    const int gate = wave >> 1;
    const int lane = tid & 31;
    const int half = lane >> 4;
    const int l16  = lane & 15;
    const float* Wsg = Ws + gate * WGATE;
    float* gsg = gate_s + gate * GGATE;

    // Stage weights (zero-padding columns EMB..WSTR-1) and zero h state.
    for (int i = tid; i < 4 * EMB * WSTR; i += 256) {
        int g = i / WGATE;
        int r = i - g * WGATE;
        int k = r / WSTR;
        int n = r - k * WSTR;
        const float* Wg = (g == 0 ? Wfh : g == 1 ? Wih : g == 2 ? Wch : Woh)
                          + layer * EMB * EMB;
        Ws[i] = (n < EMB) ? Wg[k * EMB + n] : 0.0f;
    }
    for (int e = tid; e < 16 * GSTR; e += 256) h_s[e] = 0.0f;
    float c_reg[7];
#pragma unroll
    for (int j = 0; j < 7; ++j) c_reg[j] = 0.0f;
    __syncthreads();

    for (int t = 0; t < TSTEPS; ++t) {
        // phase 1 (wave-uniform branch; EXEC all-ones inside each path)
        if (wave & 1)
            rec_phase1<3>(xpart, Wsg, h_s, gsg, 64, b0, t, gate, l16, half);
        else
            rec_phase1<4>(xpart, Wsg, h_s, gsg, 0, b0, t, gate, l16, half);
        __syncthreads();
        // phase 2: elementwise cell update; h back to LDS + global
#pragma unroll
        for (int j = 0; j < 7; ++j) {
            int e = tid + j * 256;
            if (e < 16 * EMB) {
                int m = e / EMB;
                int n = e - m * EMB;
                float ft = sigmoidf_(gate_s[0 * GGATE + m * GSTR + n]);
                float it = sigmoidf_(gate_s[1 * GGATE + m * GSTR + n]);
                float ct = tanhf(gate_s[2 * GGATE + m * GSTR + n]);
                float ot = sigmoidf_(gate_s[3 * GGATE + m * GSTR + n]);
                float c  = ft * c_reg[j] + it * ct;
                float h  = ot * tanhf(c);
                c_reg[j] = c;
                h_s[m * GSTR + n] = h;
                hseq[((size_t)(b0 + m) * TSTEPS + t) * EMB + n] = h;
            }
        }
        __syncthreads();
    }
}

// ----------------------------------------- out = h_last @ Wfinal + bfinal
// M = 2048, K = 100, N = 50257.  Block: 32 M-rows x 256 N.  kc-outer with
// 8 accumulators live: each kc = clause of 8 B-loads + 8 independent WMMAs.
__global__ __launch_bounds__(128) void k_final(const float* __restrict__ hseq,
    const float* __restrict__ Wfinal, const float* __restrict__ bfinal,
    float* __restrict__ out) {
    const int m0   = blockIdx.x * 32;
    const int wave = threadIdx.x >> 5;
    const int lane = threadIdx.x & 31;
    const int half = lane >> 4;
    const int l16  = lane & 15;
    const int nbase = blockIdx.y * 256 + wave * 64;

    v2f a0[25], a1[25];
#pragma unroll
    for (int kc = 0; kc < 25; ++kc) {
        const float* p0 = hseq + ((size_t)(m0 + l16) * TSTEPS + (TSTEPS - 1)) * EMB
                        + kc * 4 + 2 * half;
        const float* p1 = p0 + (size_t)16 * TSTEPS * EMB;
        a0[kc].x = p0[0]; a0[kc].y = p0[1];
        a1[kc].x = p1[0]; a1[kc].y = p1[1];
    }

    if (nbase + 63 < VOCABN) {
        // interior: no guards anywhere (wave-uniform branch)
        v8f acc0[4], acc1[4];
#pragma unroll
        for (int nt = 0; nt < 4; ++nt) {
            float bb = bfinal[nbase + nt * 16 + l16];
#pragma unroll
            for (int r = 0; r < 8; ++r) { acc0[nt][r] = bb; acc1[nt][r] = bb; }
        }
        gemm_tiles<4>(a0, a1, Wfinal, VOCABN, nbase, l16, half, NOLIMIT, acc0, acc1);
#pragma unroll
        for (int nt = 0; nt < 4; ++nt) {
            int n = nbase + nt * 16 + l16;
#pragma unroll
            for (int r = 0; r < 8; ++r) {
                out[(size_t)(m0 + r + 8 * half) * VOCABN + n]      = acc0[nt][r];
                out[(size_t)(m0 + 16 + r + 8 * half) * VOCABN + n] = acc1[nt][r];
            }
        }
    } else {
        // edge tile: guarded loads/stores
        v8f acc0[4], acc1[4];
#pragma unroll
        for (int nt = 0; nt < 4; ++nt) {
            int n = nbase + nt * 16 + l16;
            float bb = (n < VOCABN) ? bfinal[n] : 0.0f;
#pragma unroll
            for (int r = 0; r < 8; ++r) { acc0[nt][r] = bb; acc1[nt][r] = bb; }
        }
        gemm_tiles<4>(a0, a1, Wfinal, VOCABN, nbase, l16, half, VOCABN, acc0, acc1);
#pragma unroll
        for (int nt = 0; nt < 4; ++nt) {
            int n = nbase + nt * 16 + l16;
            if (n < VOCABN) {
#pragma unroll
                for (int r = 0; r < 8; ++r) {
                    out[(size_t)(m0 + r + 8 * half) * VOCABN + n]      = acc0[nt][r];
                    out[(size_t)(m0 + 16 + r + 8 * half) * VOCABN + n] = acc1[nt][r];
                }
            }
        }
    }
}

extern "C" void kernel_launch(void* const* d_in, const int* in_sizes, int n_in,
                              void* d_out, int out_size, void* d_ws, size_t ws_size,
                              hipStream_t stream) {
    (void)in_sizes; (void)n_in; (void)out_size; (void)ws_size;
    const int*   x      = (const int*)d_in[0];
    const float* emb    = (const float*)d_in[1];
    const float* Wfh    = (const float*)d_in[2];
    const float* Wfx    = (const float*)d_in[3];
    const float* Wih    = (const float*)d_in[4];
    const float* Wix    = (const float*)d_in[5];
    const float* Wch    = (const float*)d_in[6];
    const float* Wcx    = (const float*)d_in[7];
    const float* Woh    = (const float*)d_in[8];
    const float* Wox    = (const float*)d_in[9];
    const float* bf     = (const float*)d_in[10];
    const float* bi     = (const float*)d_in[11];
    const float* bc     = (const float*)d_in[12];
    const float* bo     = (const float*)d_in[13];
    const float* Wfinal = (const float*)d_in[14];
    const float* bfinal = (const float*)d_in[15];
    float* out = (float*)d_out;

    // workspace: hseq [ROWS*EMB] | xpart [ROWS*GDIM]   (~61.5 MB total)
    float* hseq  = (float*)d_ws;
    float* xpart = hseq + ((size_t)ROWS * EMB + 64);

    k_embed<<<(ROWS * EMB + 255) / 256, 256, 0, stream>>>(x, emb, hseq);

    const size_t smem_bytes = SMEM_FLOATS * sizeof(float);   // 215360 B < 320KB/WGP
    for (int l = 0; l < STACKS; ++l) {
        k_xgemm<<<ROWS / 32, 128, 0, stream>>>(hseq, Wfx, Wix, Wcx, Wox,
                                               bf, bi, bc, bo, xpart, l);
        k_recurrent<<<BATCH / 16, 256, smem_bytes, stream>>>(xpart, Wfh, Wih, Wch,
                                                             Woh, hseq, l);
    }

    dim3 grid(BATCH / 32, (VOCABN + 255) / 256);
    k_final<<<grid, 128, 0, stream>>>(hseq, Wfinal, bfinal, out);
}